// GGCNConv_55241869361500
// MI455X (gfx1250) — compile-verified
//
#include <hip/hip_runtime.h>
#include <math.h>

#define N_NODES 50000
#define N_EDGES 1600000
#define DIM 64
#define EPS 1e-5f

typedef float v2f __attribute__((ext_vector_type(2)));
typedef float v8f __attribute__((ext_vector_type(8)));

// D = A(16x4) * B(4x16) + C, f32, wave32 WMMA
__device__ __forceinline__ v8f wmma_f32_16x16x4(v2f a, v2f b, v8f c) {
    return __builtin_amdgcn_wmma_f32_16x16x4_f32(
        /*neg_a=*/false, a, /*neg_b=*/false, b,
        /*c_mod=*/(short)0, c, /*reuse_a=*/false, /*reuse_b=*/false);
}

// ---------------------------------------------------------------------------
// Kernel 0: zero the stats accumulators (esum, esq, nsum, nsq -> 4*64 floats)
// ---------------------------------------------------------------------------
__global__ void zero_kernel(float* __restrict__ p, int n) {
    int i = blockIdx.x * blockDim.x + threadIdx.x;
    if (i < n) p[i] = 0.0f;
}

// ---------------------------------------------------------------------------
// Kernel 1: node-side linears via WMMA.
//   xg1 = x@Wg[0:64], xg2 = x@Wg[64:128], hacc = x@Ws+bs, dstl = x@Wd+bd
// One wave computes a 16-node x 64-col tile; 8 waves per block.
// ---------------------------------------------------------------------------
__global__ __launch_bounds__(256) void node_linear_kernel(
    const float* __restrict__ x,
    const float* __restrict__ Wg,                       // [192,64] row-major
    const float* __restrict__ Ws, const float* __restrict__ bs,
    const float* __restrict__ Wd, const float* __restrict__ bd,
    float* __restrict__ xg1, float* __restrict__ xg2,
    float* __restrict__ dstl, float* __restrict__ hacc)
{
    __shared__ float w[4][DIM * DIM];                   // 64 KB
    {
        const float* srcs[4] = {Wg, Wg + DIM * DIM, Ws, Wd};
        for (int wi = 0; wi < 4; ++wi)
            for (int i = threadIdx.x; i < DIM * DIM / 4; i += blockDim.x)
                ((float4*)w[wi])[i] = ((const float4*)srcs[wi])[i];
    }
    __syncthreads();

    const int lane = threadIdx.x & 31;
    const int wave = threadIdx.x >> 5;
    const int lr   = lane & 15;
    const int half = lane >> 4;

    const int base = (blockIdx.x * 8 + wave) * 16;
    if (base >= N_NODES) return;                        // wave-uniform

    // A fragments: lane holds row (base+lr), elements {4k+2*half, +1}
    v2f a[16];
    const float* xrow = x + (size_t)(base + lr) * DIM + 2 * half;
#pragma unroll
    for (int k = 0; k < 16; ++k) a[k] = *(const v2f*)(xrow + 4 * k);

    float* outs[4]         = {xg1, xg2, hacc, dstl};
    const float* biases[4] = {nullptr, nullptr, bs, bd};

    for (int wi = 0; wi < 4; ++wi) {
        const float* wm = w[wi];
        float* op       = outs[wi];
        const float* bp = biases[wi];
        for (int t = 0; t < 4; ++t) {
            const int col = t * 16 + lr;
            v8f acc = {0.f, 0.f, 0.f, 0.f, 0.f, 0.f, 0.f, 0.f};
#pragma unroll
            for (int k = 0; k < 16; ++k) {
                const int r0 = 4 * k + 2 * half;
                v2f b;
                b.x = wm[r0 * DIM + col];
                b.y = wm[(r0 + 1) * DIM + col];
                acc = wmma_f32_16x16x4(a[k], b, acc);
            }
            const float bias = bp ? bp[col] : 0.0f;
#pragma unroll
            for (int r = 0; r < 8; ++r) {
                const int row = base + r + 8 * half;    // C layout: VGPR r
                op[(size_t)row * DIM + col] = acc[r] + bias;
            }
        }
    }
}

// ---------------------------------------------------------------------------
// Kernel 2: edge pass. Per 16-edge wave tile:
//   m = edge_attr@Wg3 (WMMA) + xg1[src] + xg2[dst] + bg
//   -> store m to edge_out region; accumulate col sums (batchnorm stats);
//   -> atomicAdd(sigma(m/8) * dstl[dst]) into hacc[src]  (segment_sum)
// ---------------------------------------------------------------------------
__global__ __launch_bounds__(256) void edge_kernel(
    const float* __restrict__ edge_attr,
    const int*   __restrict__ eidx,                     // [2*E] (src | dst)
    const float* __restrict__ Wg,                       // rows 128..191 used
    const float* __restrict__ bg,
    const float* __restrict__ xg1, const float* __restrict__ xg2,
    const float* __restrict__ dstl,
    float* __restrict__ hacc,
    float* __restrict__ m_out,                          // d_out + N*DIM
    float* __restrict__ esum, float* __restrict__ esq)
{
    __shared__ float w3[DIM * DIM];                     // 16 KB
    __shared__ float ssum[DIM];
    __shared__ float ssq[DIM];

    const float* Wg3 = Wg + 2 * DIM * DIM;
    for (int i = threadIdx.x; i < DIM * DIM / 4; i += blockDim.x)
        ((float4*)w3)[i] = ((const float4*)Wg3)[i];
    if (threadIdx.x < DIM) { ssum[threadIdx.x] = 0.f; ssq[threadIdx.x] = 0.f; }
    __syncthreads();

    const int lane = threadIdx.x & 31;
    const int wave = threadIdx.x >> 5;
    const int lr   = lane & 15;
    const int half = lane >> 4;

    const long long base = (long long)(blockIdx.x * 8 + wave) * 16;
    const bool active = (base < N_EDGES);               // wave-uniform

    if (active) {
        v2f a[16];
        const float* erow = edge_attr + (size_t)(base + lr) * DIM + 2 * half;
#pragma unroll
        for (int k = 0; k < 16; ++k) a[k] = *(const v2f*)(erow + 4 * k);

        int sidx[8], didx[8];
#pragma unroll
        for (int r = 0; r < 8; ++r) {
            const long long row = base + r + 8 * half;
            sidx[r] = eidx[row];
            didx[r] = eidx[N_EDGES + row];
        }

        for (int t = 0; t < 4; ++t) {
            const int col = t * 16 + lr;
            v8f acc = {0.f, 0.f, 0.f, 0.f, 0.f, 0.f, 0.f, 0.f};
#pragma unroll
            for (int k = 0; k < 16; ++k) {
                const int r0 = 4 * k + 2 * half;
                v2f b;
                b.x = w3[r0 * DIM + col];
                b.y = w3[(r0 + 1) * DIM + col];
                acc = wmma_f32_16x16x4(a[k], b, acc);
            }
            const float bgc = bg[col];
            float lsum = 0.f, lsq = 0.f;
#pragma unroll
            for (int r = 0; r < 8; ++r) {
                const long long row = base + r + 8 * half;
                const float m = acc[r]
                              + xg1[(size_t)sidx[r] * DIM + col]
                              + xg2[(size_t)didx[r] * DIM + col] + bgc;
                m_out[(size_t)row * DIM + col] = m;     // saved for pass 2
                lsum += m; lsq += m * m;
                const float sg = 1.0f / (1.0f + __expf(-m * 0.125f)); // /sqrt(64)
                const float dv = dstl[(size_t)didx[r] * DIM + col];
                atomicAdd(&hacc[(size_t)sidx[r] * DIM + col], sg * dv);
            }
            atomicAdd(&ssum[col], lsum);
            atomicAdd(&ssq[col], lsq);
        }
    }
    __syncthreads();
    if (threadIdx.x < DIM) {
        atomicAdd(&esum[threadIdx.x], ssum[threadIdx.x]);
        atomicAdd(&esq[threadIdx.x],  ssq[threadIdx.x]);
    }
}

// ---------------------------------------------------------------------------
// Kernel 3: per-column sum / sumsq of hacc over N (node batchnorm stats)
// ---------------------------------------------------------------------------
__global__ __launch_bounds__(256) void node_stats_kernel(
    const float* __restrict__ hacc,
    float* __restrict__ nsum, float* __restrict__ nsq)
{
    __shared__ float ss[256], sq[256];
    const int col = threadIdx.x & 63;
    const int grp = threadIdx.x >> 6;                   // 0..3
    float s = 0.f, q = 0.f;
    for (int row = blockIdx.x * 4 + grp; row < N_NODES; row += gridDim.x * 4) {
        const float v = hacc[(size_t)row * DIM + col];
        s += v; q += v * v;
    }
    ss[threadIdx.x] = s; sq[threadIdx.x] = q;
    __syncthreads();
    if (grp == 0) {
        s = ss[col] + ss[col + 64] + ss[col + 128] + ss[col + 192];
        q = sq[col] + sq[col + 64] + sq[col + 128] + sq[col + 192];
        atomicAdd(&nsum[col], s);
        atomicAdd(&nsq[col],  q);
    }
}

// ---------------------------------------------------------------------------
// Kernel 4: finalize stats -> {mean, rsqrt(var+eps)} for nodes and edges
// ---------------------------------------------------------------------------
__global__ void finalize_stats_kernel(
    const float* __restrict__ nsum, const float* __restrict__ nsq,
    const float* __restrict__ esum, const float* __restrict__ esq,
    float* __restrict__ nstat, float* __restrict__ estat)
{
    const int c = threadIdx.x;
    if (c < DIM) {
        const float nm = nsum[c] / (float)N_NODES;
        const float nv = nsq[c] / (float)N_NODES - nm * nm;
        nstat[c] = nm; nstat[DIM + c] = rsqrtf(nv + EPS);
        const float em = esum[c] / (float)N_EDGES;
        const float ev = esq[c] / (float)N_EDGES - em * em;
        estat[c] = em; estat[DIM + c] = rsqrtf(ev + EPS);
    }
}

// ---------------------------------------------------------------------------
// Kernel 5: node_out = relu(batchnorm(hacc))
// ---------------------------------------------------------------------------
__global__ __launch_bounds__(256) void node_norm_kernel(
    const float* __restrict__ hacc, const float* __restrict__ nstat,
    const float* __restrict__ gamma, const float* __restrict__ beta,
    float* __restrict__ out)
{
    const size_t total = (size_t)N_NODES * DIM;
    for (size_t i = (size_t)blockIdx.x * blockDim.x + threadIdx.x; i < total;
         i += (size_t)gridDim.x * blockDim.x) {
        const int c = (int)(i & 63);
        const float v = (hacc[i] - nstat[c]) * nstat[DIM + c] * gamma[c] + beta[c];
        out[i] = v > 0.f ? v : 0.f;
    }
}

// ---------------------------------------------------------------------------
// Kernel 6: edge_out = relu(batchnorm(m))  (in place on d_out edge region)
// ---------------------------------------------------------------------------
__global__ __launch_bounds__(256) void edge_norm_kernel(
    float* __restrict__ m_io, const float* __restrict__ estat,
    const float* __restrict__ gamma, const float* __restrict__ beta)
{
    const size_t total = (size_t)N_EDGES * DIM;
    for (size_t i = (size_t)blockIdx.x * blockDim.x + threadIdx.x; i < total;
         i += (size_t)gridDim.x * blockDim.x) {
        const int c = (int)(i & 63);
        const float v = (m_io[i] - estat[c]) * estat[DIM + c] * gamma[c] + beta[c];
        m_io[i] = v > 0.f ? v : 0.f;
    }
}

// ---------------------------------------------------------------------------
extern "C" void kernel_launch(void* const* d_in, const int* in_sizes, int n_in,
                              void* d_out, int out_size, void* d_ws, size_t ws_size,
                              hipStream_t stream) {
    const float* x         = (const float*)d_in[0];
    const int*   edge_idx  = (const int*)  d_in[1];   // [2,E]
    const float* edge_attr = (const float*)d_in[2];
    const float* Wg        = (const float*)d_in[3];   // [192,64]
    const float* bg        = (const float*)d_in[4];
    const float* Ws        = (const float*)d_in[5];
    const float* bs        = (const float*)d_in[6];
    const float* Wd        = (const float*)d_in[7];
    const float* bd        = (const float*)d_in[8];
    const float* gamma_n   = (const float*)d_in[9];
    const float* beta_n    = (const float*)d_in[10];
    const float* gamma_e   = (const float*)d_in[11];
    const float* beta_e    = (const float*)d_in[12];

    float* out      = (float*)d_out;
    float* node_out = out;                               // [N,64]
    float* edge_out = out + (size_t)N_NODES * DIM;       // [E,64] (holds m in pass 1)

    // Workspace layout (floats)
    const size_t NB = (size_t)N_NODES * DIM;             // 3.2M floats
    float* ws   = (float*)d_ws;
    float* xg1  = ws;                                    // x @ Wg1
    float* xg2  = ws + NB;                               // x @ Wg2
    float* dstl = ws + 2 * NB;                           // x @ Wd + bd
    float* hacc = ws + 3 * NB;                           // src-linear + segment_sum
    float* stat = ws + 4 * NB;
    float* esum  = stat;                                 // 64
    float* esq   = stat + 64;                            // 64
    float* nsum  = stat + 128;                           // 64
    float* nsq   = stat + 192;                           // 64
    float* nstat = stat + 256;                           // mean|istd : 128
    float* estat = stat + 384;                           // mean|istd : 128

    // 0) zero stat accumulators
    zero_kernel<<<1, 256, 0, stream>>>(stat, 256);

    // 1) node-side linears (WMMA): 3125 tiles of 16 nodes, 8 waves/block
    node_linear_kernel<<<(3125 + 7) / 8, 256, 0, stream>>>(
        x, Wg, Ws, bs, Wd, bd, xg1, xg2, dstl, hacc);

    // 2) edge pass (WMMA): 100000 tiles of 16 edges, 8 waves/block = 12500 blocks
    edge_kernel<<<12500, 256, 0, stream>>>(
        edge_attr, edge_idx, Wg, bg, xg1, xg2, dstl, hacc, edge_out, esum, esq);

    // 3) node batchnorm stats
    node_stats_kernel<<<256, 256, 0, stream>>>(hacc, nsum, nsq);

    // 4) finalize mean / inv-std
    finalize_stats_kernel<<<1, 64, 0, stream>>>(nsum, nsq, esum, esq, nstat, estat);

    // 5) node_out = relu(bn(h))
    node_norm_kernel<<<4096, 256, 0, stream>>>(hacc, nstat, gamma_n, beta_n, node_out);

    // 6) edge_out = relu(bn(m)) in place
    edge_norm_kernel<<<8192, 256, 0, stream>>>(edge_out, estat, gamma_e, beta_e);

    (void)in_sizes; (void)n_in; (void)out_size; (void)ws_size;
}